// SSD_27032524161323
// MI455X (gfx1250) — compile-verified
//
#include <hip/hip_runtime.h>
#include <hip/hip_bf16.h>
#include <stdint.h>

#define NUM_CLS 21
#define NB 32
#define NP 8732
#define KK 400            // pre-NMS top-k
#define NTOPK 200         // output top-k
#define CONF_TH 0.01f
#define NMS_TH 0.45f
#define VAR0 0.1f
#define VAR1 0.2f
#define NTILE 25          // 400/16
#define NKW 13            // ceil(400/32) keep/adj words per row
#define SORTN 512

typedef __attribute__((ext_vector_type(2))) float v2f;
typedef __attribute__((ext_vector_type(8))) float v8f;

#if __has_builtin(__builtin_amdgcn_global_load_async_to_lds_b32)
#define HAVE_ASYNC_LDS 1
#else
#define HAVE_ASYNC_LDS 0
#endif

typedef __attribute__((address_space(1))) int gint_t;
typedef __attribute__((address_space(3))) int lint_t;

__device__ __forceinline__ unsigned long long mk_key(float s, int i) {
  // softmax scores are positive -> uint bit order == float order.
  // low 32 bits = ~index: among equal scores, smaller index sorts higher
  // (matches jax.lax.top_k stable tie-break). All keys distinct.
  return ((unsigned long long)__float_as_uint(s) << 32) | (unsigned)(~(unsigned)i);
}

// ---------------------------------------------------------------------------
// Kernel 1 (optional, needs 23.5MB workspace): softmax + transpose to [B,C,P]
// ---------------------------------------------------------------------------
__global__ void __launch_bounds__(256)
ssd_softmax_T(const float* __restrict__ conf, float* __restrict__ scoresT) {
  int t = blockIdx.x * blockDim.x + threadIdx.x;
  if (t >= NB * NP) return;
  int b = t / NP, p = t % NP;
  const float* c = conf + (size_t)t * NUM_CLS;
  float m = c[0];
#pragma unroll
  for (int i = 1; i < NUM_CLS; ++i) m = fmaxf(m, c[i]);
  float e[NUM_CLS];
  float s = 0.f;
#pragma unroll
  for (int i = 0; i < NUM_CLS; ++i) { e[i] = expf(c[i] - m); s += e[i]; }
  float inv = 1.f / s;
#pragma unroll
  for (int i = 0; i < NUM_CLS; ++i)
    scoresT[((size_t)(b * NUM_CLS + i)) * NP + p] = e[i] * inv;
}

// ---------------------------------------------------------------------------
// Kernel 2: one block per (batch, class): top-400 select -> sort -> WMMA IoU
// adjacency -> greedy NMS on bitmasks -> stable compaction to 200 outputs.
// ---------------------------------------------------------------------------
__global__ void __launch_bounds__(256)
ssd_nms(const float* __restrict__ loc, const float* __restrict__ conf,
        const float* __restrict__ priors, const float* __restrict__ scoresT,
        float* __restrict__ out) {
  const int bc  = blockIdx.x;
  const int b   = bc / NUM_CLS;
  const int c   = bc % NUM_CLS;
  const int tid = threadIdx.x;
  float* outp = out + (size_t)bc * NTOPK * 5;

  if (c == 0) {  // background class: reference zeroes it
    for (int i = tid; i < NTOPK * 5; i += 256) outp[i] = 0.f;
    return;
  }

  __shared__ float s_scores[NP];                 // 34928 B (overlaid below)
  __shared__ unsigned long long s_sort[SORTN];   // 4096 B
  __shared__ unsigned s_hist[256];               // 1024 B
  __shared__ unsigned long long s_prefkey;
  __shared__ int s_need;
  __shared__ int s_cnt;

  // ---- overlay of s_scores region (dead after gather phase) ----
  float*    s_box  = (float*)s_scores;            // [KK*4]  6400 B
  float*    s_sc   = s_box + KK * 4;              // [KK]    1600 B
  float*    s_area = s_sc + KK;                   // [KK]    1600 B
  unsigned* s_adj  = (unsigned*)(s_area + KK);    // [KK*13] 20800 B
  unsigned* s_keep = s_adj + KK * NKW;            // [13]
  unsigned* s_rpref = s_keep + NKW;               // [13]

  // ================= Phase 1: stage per-class scores into LDS ==============
  if (scoresT) {
    const float* g = scoresT + (size_t)bc * NP;
#if HAVE_ASYNC_LDS
    for (int i = tid; i < NP; i += 256) {
      __builtin_amdgcn_global_load_async_to_lds_b32(
          (gint_t*)(uintptr_t)(g + i),
          (lint_t*)(uintptr_t)(&s_scores[i]),
          0, 0);
    }
#if __has_builtin(__builtin_amdgcn_s_wait_asynccnt)
    __builtin_amdgcn_s_wait_asynccnt(0);
#else
    asm volatile("s_wait_asynccnt 0" ::: "memory");
#endif
#else
    for (int i = tid; i < NP; i += 256) s_scores[i] = g[i];
#endif
  } else {
    // no workspace: recompute softmax for this class on the fly
    for (int i = tid; i < NP; i += 256) {
      const float* cc = conf + ((size_t)b * NP + i) * NUM_CLS;
      float m = cc[0];
#pragma unroll
      for (int k = 1; k < NUM_CLS; ++k) m = fmaxf(m, cc[k]);
      float s = 0.f, ec = 0.f;
#pragma unroll
      for (int k = 0; k < NUM_CLS; ++k) {
        float e = expf(cc[k] - m);
        s += e;
        if (k == c) ec = e;
      }
      s_scores[i] = ec / s;
    }
  }
  if (tid == 0) { s_prefkey = 0ull; s_need = KK; }
  __syncthreads();

  // ========== Phase 2: exact top-400 threshold via 64-bit radix select =====
  for (int d = 7; d >= 0; --d) {
    s_hist[tid] = 0u;
    __syncthreads();
    unsigned long long pref  = s_prefkey;
    unsigned long long hmask = (d == 7) ? 0ull : (~0ull << ((d + 1) * 8));
    for (int i = tid; i < NP; i += 256) {
      unsigned long long key = mk_key(s_scores[i], i);
      if ((key & hmask) == pref) {
        unsigned bkt = (unsigned)(key >> (d * 8)) & 255u;
        atomicAdd(&s_hist[bkt], 1u);
      }
    }
    __syncthreads();
    if (tid == 0) {
      int need = s_need;
      unsigned acc = 0;
      int q = 0;
      for (int bb = 255; bb >= 0; --bb) {
        unsigned cc2 = s_hist[bb];
        if (acc + cc2 >= (unsigned)need) { q = bb; break; }
        acc += cc2;
      }
      s_need = need - (int)acc;
      s_prefkey = pref | ((unsigned long long)q << (d * 8));
    }
    __syncthreads();
  }

  // gather exactly the 400 keys >= threshold (keys are distinct)
  if (tid == 0) s_cnt = 0;
  __syncthreads();
  {
    unsigned long long thr = s_prefkey;
    for (int i = tid; i < NP; i += 256) {
      unsigned long long key = mk_key(s_scores[i], i);
      if (key >= thr) {
        int pos = atomicAdd(&s_cnt, 1);
        if (pos < SORTN) s_sort[pos] = key;
      }
    }
  }
  __syncthreads();
  {
    int cnt = s_cnt;
    for (int i = tid; i < SORTN; i += 256)
      if (i >= cnt) s_sort[i] = 0ull;
  }

  // ================= Phase 3: bitonic sort (descending, 512) ===============
  for (unsigned k2 = 2; k2 <= SORTN; k2 <<= 1) {
    for (unsigned j2 = k2 >> 1; j2 > 0; j2 >>= 1) {
      __syncthreads();
      for (unsigned i2 = tid; i2 < SORTN; i2 += 256) {
        unsigned ix = i2 ^ j2;
        if (ix > i2) {
          unsigned long long a = s_sort[i2], bb = s_sort[ix];
          bool asc = (i2 & k2) != 0;  // descending overall
          bool sw = asc ? (a > bb) : (a < bb);
          if (sw) { s_sort[i2] = bb; s_sort[ix] = a; }
        }
      }
    }
  }
  __syncthreads();

  // ========== Phase 4: extract top-400, decode boxes, areas (overlay) ======
  for (int r = tid; r < KK; r += 256) {
    unsigned long long key = s_sort[r];
    float sc = __uint_as_float((unsigned)(key >> 32));
    unsigned idx = ~(unsigned)(key & 0xFFFFFFFFu);
    if (idx >= NP) idx = 0;  // defensive (padding can't reach top-400)
    const float* pr = priors + (size_t)idx * 4;
    const float* ll = loc + ((size_t)b * NP + idx) * 4;
    float pw = pr[2], ph = pr[3];
    float cx = pr[0] + ll[0] * VAR0 * pw;
    float cy = pr[1] + ll[1] * VAR0 * ph;
    float w = pw * expf(ll[2] * VAR1);
    float h = ph * expf(ll[3] * VAR1);
    float x1 = cx - 0.5f * w, y1 = cy - 0.5f * h;
    float x2 = cx + 0.5f * w, y2 = cy + 0.5f * h;
    s_box[r * 4 + 0] = x1; s_box[r * 4 + 1] = y1;
    s_box[r * 4 + 2] = x2; s_box[r * 4 + 3] = y2;
    s_sc[r] = sc;
    s_area[r] = (x2 - x1) * (y2 - y1);
  }
  for (int i = tid; i < KK * NKW; i += 256) s_adj[i] = 0u;
  if (tid < NKW) s_keep[tid] = 0u;
  __syncthreads();

  // valid mask (conf threshold)
  for (int r = tid; r < KK; r += 256)
    if (s_sc[r] > CONF_TH) atomicOr(&s_keep[r >> 5], 1u << (r & 31));

  // ====== Phase 5: IoU adjacency tiles — WMMA computes area_i + area_j =====
  {
    int lane = tid & 31;
    int wv = tid >> 5;
    for (int t5 = wv; t5 < NTILE * NTILE; t5 += 8) {
      int ti = t5 / NTILE, tj = t5 % NTILE;
      // A[m] = (area_m, 1, 0, 0) ; B = (1^T ; area_n^T ; 0 ; 0)
      float aA = (lane < 16) ? s_area[ti * 16 + lane] : 0.f;
      float bA = (lane < 16) ? s_area[tj * 16 + lane] : 0.f;
      v2f A;  A.x  = aA;                       A.y  = (lane < 16) ? 1.f : 0.f;
      v2f Bv; Bv.x = (lane < 16) ? 1.f : 0.f;  Bv.y = bA;
      v8f C = {0.f, 0.f, 0.f, 0.f, 0.f, 0.f, 0.f, 0.f};
      // D[m][n] = area_i[m] + area_j[n]  (rank-2 outer product on matrix pipe)
      v8f D = __builtin_amdgcn_wmma_f32_16x16x4_f32(
          false, A, false, Bv, (short)0, C, false, false);
      int n    = lane & 15;
      int mh   = (lane >= 16) ? 8 : 0;
      bool wr  = (lane & 15) == 0;           // lanes 0 and 16 write rows
      unsigned sh = (unsigned)(tj & 1) * 16u;
#pragma unroll
      for (int v = 0; v < 8; ++v) {
        int i = ti * 16 + v + mh;
        int j = tj * 16 + n;
        const float* bi = &s_box[i * 4];
        const float* bj = &s_box[j * 4];
        float lx = fmaxf(bi[0], bj[0]);
        float ly = fmaxf(bi[1], bj[1]);
        float rx = fminf(bi[2], bj[2]);
        float ry = fminf(bi[3], bj[3]);
        float inter = fmaxf(rx - lx, 0.f) * fmaxf(ry - ly, 0.f);
        float uni = D[v] - inter;                 // union from WMMA tile
        // iou > T  <=>  inter > T * max(union, eps)   (denominator > 0)
        bool pred = inter > NMS_TH * fmaxf(uni, 1e-12f);
#if __has_builtin(__builtin_amdgcn_ballot_w32)
        unsigned msk = __builtin_amdgcn_ballot_w32(pred);
#else
        unsigned msk = (unsigned)__ballot(pred);
#endif
        // bits[15:0] -> row ti*16+v (lane 0), bits[31:16] -> row +8 (lane 16)
        unsigned rowbits = (lane == 0) ? (msk & 0xFFFFu) : (msk >> 16);
        if (wr)
          atomicOr(&s_adj[i * NKW + (tj >> 1)], rowbits << sh);
      }
    }
  }
  __syncthreads();

  // ====== Phase 6: greedy NMS — sequential, but only 13-word mask ops ======
  for (int i = 0; i < KK; ++i) {
    unsigned kw = s_keep[i >> 5];
    bool alive = (kw >> (i & 31)) & 1u;
    __syncthreads();
    if (alive && tid < NKW) {
      int wi = i >> 5, bp = i & 31;
      unsigned gt;  // mask of j > i within this word
      if (tid < wi)       gt = 0u;
      else if (tid == wi) gt = (bp == 31) ? 0u : (0xFFFFFFFFu << (bp + 1));
      else                gt = 0xFFFFFFFFu;
      s_keep[tid] &= ~(s_adj[i * NKW + tid] & gt);
    }
    __syncthreads();
  }

  // ====== Phase 7: stable compaction to front, zeros elsewhere =============
  if (tid == 0) {
    unsigned run = 0;
    for (int w = 0; w < NKW; ++w) { s_rpref[w] = run; run += __popc(s_keep[w]); }
  }
  for (int i = tid; i < NTOPK * 5; i += 256) outp[i] = 0.f;
  __syncthreads();
  for (int r = tid; r < KK; r += 256) {
    unsigned w = s_keep[r >> 5];
    int bp = r & 31;
    if ((w >> bp) & 1u) {
      int rank = (int)s_rpref[r >> 5] + __popc(w & ((1u << bp) - 1u));
      if (rank < NTOPK) {
        float* o = outp + rank * 5;
        o[0] = s_sc[r];
        o[1] = s_box[r * 4 + 0];
        o[2] = s_box[r * 4 + 1];
        o[3] = s_box[r * 4 + 2];
        o[4] = s_box[r * 4 + 3];
      }
    }
  }
}

// ---------------------------------------------------------------------------
extern "C" void kernel_launch(void* const* d_in, const int* in_sizes, int n_in,
                              void* d_out, int out_size, void* d_ws, size_t ws_size,
                              hipStream_t stream) {
  const float* locations   = (const float*)d_in[0];  // [B,P,4]
  const float* confidences = (const float*)d_in[1];  // [B,P,21]
  const float* priors      = (const float*)d_in[2];  // [P,4]
  float* out = (float*)d_out;                        // [B,21,200,5]

  const size_t needWS = (size_t)NB * NUM_CLS * NP * sizeof(float);
  float* scoresT = nullptr;
  if (ws_size >= needWS) {
    scoresT = (float*)d_ws;
    int n = NB * NP;
    hipLaunchKernelGGL(ssd_softmax_T, dim3((n + 255) / 256), dim3(256), 0, stream,
                       confidences, scoresT);
  }
  hipLaunchKernelGGL(ssd_nms, dim3(NB * NUM_CLS), dim3(256), 0, stream,
                     locations, confidences, priors, scoresT, out);
}